// CausalRefineNet_83983790506180
// MI455X (gfx1250) — compile-verified
//
#include <hip/hip_runtime.h>
#include <hip/hip_bf16.h>
#include <math.h>

typedef __attribute__((ext_vector_type(2))) float v2f;
typedef __attribute__((ext_vector_type(8))) float v8f;
typedef int v4i __attribute__((vector_size(16)));
typedef __attribute__((address_space(1))) v4i* gv4p;   // global-AS pointer to int4
typedef __attribute__((address_space(3))) v4i* lv4p;   // LDS-AS pointer to int4

#define NPTS   8192
#define NBATCH 4
#define KNN    20
#define CHUNK  2048
#define WAVES  4
#define BLOCK  (WAVES * 32)

#if defined(__has_builtin)
#if __has_builtin(__builtin_amdgcn_global_load_async_to_lds_b128) && \
    __has_builtin(__builtin_amdgcn_s_wait_asynccnt)
#define USE_ASYNC_LDS 1
#endif
#endif

__global__ __launch_bounds__(BLOCK)
void knn_normals_wmma(const float* __restrict__ pts_all,
                      float* __restrict__ out)
{
    __shared__ __align__(16) float s_raw[CHUNK * 3];        // 24 KB: packed candidate xyz
    __shared__ __align__(16) float s_dist[WAVES][32][16];   // 8 KB : staged WMMA tiles

    const int b    = blockIdx.y;
    const int tid  = threadIdx.x;
    const int wave = tid >> 5;
    const int lane = tid & 31;
    const int lo   = lane & 15;
    const int hi   = lane >> 4;               // 0: lanes 0-15, 1: lanes 16-31

    const float* __restrict__ P = pts_all + (size_t)b * NPTS * 3;

    const int qIdx = blockIdx.x * BLOCK + tid;            // query owned by this lane
    const float qx = P[qIdx*3+0], qy = P[qIdx*3+1], qz = P[qIdx*3+2];
    const float qsq = qx*qx + qy*qy + qz*qz;

    // ---- two A tiles for this wave (queries qw..qw+31) ---------------------
    // f32 16x4 A layout: lanes 0-15 hold (K0,K1) of row M=lane; lanes 16-31 (K2,K3).
    const int qw = blockIdx.x * BLOCK + wave * 32;
    const int m0 = qw + lo;
    const int m1 = qw + 16 + lo;
    v2f A0, A1;
    {
        float x0 = P[m0*3+0], y0 = P[m0*3+1], z0 = P[m0*3+2];
        float x1 = P[m1*3+0], y1 = P[m1*3+1], z1 = P[m1*3+2];
        A0.x = hi ? z0 : x0;   A0.y = hi ? 0.0f : y0;    // K3 padded with 0
        A1.x = hi ? z1 : x1;   A1.y = hi ? 0.0f : y1;
    }

    // ---- top-K list (self excluded by index) -------------------------------
    float bd[KNN];
    int   bi[KNN];
#pragma unroll
    for (int j = 0; j < KNN; ++j) { bd[j] = 3.4e38f; bi[j] = 0; }

#pragma unroll 1
    for (int cbase = 0; cbase < NPTS; cbase += CHUNK) {
        __syncthreads();                      // previous chunk's readers done
        {
            const float* Pc = P + (size_t)cbase * 3;
#ifdef USE_ASYNC_LDS
            // gfx1250 async copy: global -> LDS without VGPR bounce (ASYNCcnt)
#pragma unroll 1
            for (int i = tid; i < (CHUNK * 3) / 4; i += BLOCK) {
                __builtin_amdgcn_global_load_async_to_lds_b128(
                    (gv4p)(Pc + i * 4),
                    (lv4p)(s_raw + i * 4),
                    0, 0);
            }
            __builtin_amdgcn_s_wait_asynccnt(0);
#else
#pragma unroll 1
            for (int i = tid; i < CHUNK * 3; i += BLOCK) s_raw[i] = Pc[i];
#endif
        }
        __syncthreads();

#pragma unroll 1
        for (int t = 0; t < CHUNK / 16; ++t) {
            // B tile: candidates cbase+t*16 .. +15, column N = lo
            const int cc = t * 16 + lo;
            float cx = s_raw[cc*3+0], cy = s_raw[cc*3+1], cz = s_raw[cc*3+2];
            float csq = cx*cx + cy*cy + cz*cz;
            v2f Bt;
            Bt.x = hi ? cz : cx;              // K2 : K0
            Bt.y = hi ? 0.0f : cy;            // K3 pad (x A-pad 0) : K1
            v8f C0 = {};
            v8f C1 = {};
            C0 = __builtin_amdgcn_wmma_f32_16x16x4_f32(false, A0, false, Bt,
                                                       (short)0, C0, false, false);
            C1 = __builtin_amdgcn_wmma_f32_16x16x4_f32(false, A1, false, Bt,
                                                       (short)0, C1, false, false);
            // stage (|c|^2 - 2*dot); this lane's column is N=lo
#pragma unroll
            for (int r = 0; r < 8; ++r) {
                int M = r + 8 * hi;
                s_dist[wave][M][lo]      = csq - 2.0f * C0[r];
                s_dist[wave][16 + M][lo] = csq - 2.0f * C1[r];
            }
            __builtin_amdgcn_wave_barrier();   // same-wave LDS is hw-ordered
            // read my query's whole row with 4x b128, then register-only top-K
            const float4* rowp = (const float4*)(&s_dist[wave][lane][0]);
            float4 r0 = rowp[0], r1 = rowp[1], r2 = rowp[2], r3 = rowp[3];
            float dv[16] = { r0.x, r0.y, r0.z, r0.w,  r1.x, r1.y, r1.z, r1.w,
                             r2.x, r2.y, r2.z, r2.w,  r3.x, r3.y, r3.z, r3.w };
#pragma unroll
            for (int j = 0; j < 16; ++j) {
                float d = qsq + dv[j];
                int ci = cbase + t * 16 + j;
                if (ci != qIdx && d < bd[KNN-1]) {
                    float dd = d; int ii = ci;
#pragma unroll
                    for (int u = 0; u < KNN; ++u) {
                        bool sw = dd < bd[u];
                        float tf = sw ? bd[u] : dd;
                        bd[u]    = sw ? dd    : bd[u];
                        dd = tf;
                        int ti = sw ? bi[u] : ii;
                        bi[u]  = sw ? ii    : bi[u];
                        ii = ti;
                    }
                }
            }
            __builtin_amdgcn_wave_barrier();
        }
    }

    // ---- covariance of neighbours (query point cancels after centering) ----
    float sx = 0.f, sy = 0.f, sz = 0.f;
#pragma unroll
    for (int j = 0; j < KNN; ++j) {
        int n = bi[j];
        sx += P[n*3+0]; sy += P[n*3+1]; sz += P[n*3+2];
    }
    const float inv_k = 1.0f / (float)KNN;
    const float mx = sx*inv_k, my = sy*inv_k, mz = sz*inv_k;
    float cxx=0.f, cxy=0.f, cxz=0.f, cyy=0.f, cyz=0.f, czz=0.f;
#pragma unroll
    for (int j = 0; j < KNN; ++j) {
        int n = bi[j];
        float dx = P[n*3+0]-mx, dy = P[n*3+1]-my, dz = P[n*3+2]-mz;
        cxx += dx*dx; cxy += dx*dy; cxz += dx*dz;
        cyy += dy*dy; cyz += dy*dz; czz += dz*dz;
    }

    // ---- closed-form symmetric 3x3 eigensolve ------------------------------
    const float tr  = cxx + cyy + czz;
    const float qq  = tr * (1.0f/3.0f);
    const float p1  = cxy*cxy + cxz*cxz + cyz*cyz;
    const float axx = cxx - qq, ayy = cyy - qq, azz = czz - qq;
    const float p2  = axx*axx + ayy*ayy + azz*azz + 2.0f*p1;
    const float pp  = sqrtf(p2 * (1.0f/6.0f)) + 1e-30f;
    const float ip  = 1.0f / pp;
    const float bxx = axx*ip, bxy = cxy*ip, bxz = cxz*ip;
    const float byy = ayy*ip, byz = cyz*ip, bzz = azz*ip;
    float r = 0.5f * (bxx*(byy*bzz - byz*byz)
                    - bxy*(bxy*bzz - byz*bxz)
                    + bxz*(bxy*byz - byy*bxz));
    r = fminf(1.0f, fmaxf(-1.0f, r));
    const float phi = acosf(r) * (1.0f/3.0f);
    const float e1  = qq + 2.0f*pp*cosf(phi);                       // largest
    const float e3  = qq + 2.0f*pp*cosf(phi + 2.0943951023931953f); // smallest

    // eigenvector of e3: largest cross product of rows of (A - e3*I)
    const float m00 = cxx - e3, m11 = cyy - e3, m22 = czz - e3;
    float v0x = cxy*cyz - cxz*m11;
    float v0y = cxz*cxy - m00*cyz;
    float v0z = m00*m11 - cxy*cxy;
    float v1x = cxy*m22 - cxz*cyz;
    float v1y = cxz*cxz - m00*m22;
    float v1z = m00*cyz - cxy*cxz;
    float v2x = m11*m22 - cyz*cyz;
    float v2y = cyz*cxz - cxy*m22;
    float v2z = cxy*cyz - m11*cxz;
    const float n0 = v0x*v0x + v0y*v0y + v0z*v0z;
    const float n1 = v1x*v1x + v1y*v1y + v1z*v1z;
    const float n2 = v2x*v2x + v2y*v2y + v2z*v2z;
    float vx = v0x, vy = v0y, vz = v0z, nb = n0;
    if (n1 > nb) { vx = v1x; vy = v1y; vz = v1z; nb = n1; }
    if (n2 > nb) { vx = v2x; vy = v2y; vz = v2z; nb = n2; }

    float nl = sqrtf(vx*vx + vy*vy + vz*vz);
    float inl = 1.0f / (nl + 1e-30f);
    vx *= inl; vy *= inl; vz *= inl;
    // orient towards -p (flip if dot(n, -p) < 0), then reference's re-normalize
    const float dt = vx*(-qx) + vy*(-qy) + vz*(-qz);
    if (dt < 0.0f) { vx = -vx; vy = -vy; vz = -vz; }
    float rn = 1.0f / (sqrtf(vx*vx + vy*vy + vz*vz) + 1e-12f);
    vx *= rn; vy *= rn; vz *= rn;

    const float curv = e3 / (tr + 1e-10f);

    float* o = out + ((size_t)b * NPTS + qIdx) * 4;
    o[0] = vx; o[1] = vy; o[2] = vz; o[3] = curv;
}

extern "C" void kernel_launch(void* const* d_in, const int* in_sizes, int n_in,
                              void* d_out, int out_size, void* d_ws, size_t ws_size,
                              hipStream_t stream) {
    (void)in_sizes; (void)n_in; (void)d_ws; (void)ws_size; (void)out_size;
    const float* pts = (const float*)d_in[0];   // (4, 8192, 3) f32; d_in[1] = k (fixed 20)
    float* out = (float*)d_out;                 // (4, 8192, 4) f32
    dim3 grid(NPTS / BLOCK, NBATCH);
    knn_normals_wmma<<<grid, BLOCK, 0, stream>>>(pts, out);
}